// PowerIterationPageRank_51539607552146
// MI455X (gfx1250) — compile-verified
//
#include <hip/hip_runtime.h>
#include <stdint.h>

// PPR power iteration: H <- (1-a)*dropout(A_hat)@H + a*H0, 5 iters.
// N=100000, E=3200000, C=64. H (25.6MB) is L2-resident on MI455X (192MB L2),
// so this is L2-gather/atomic bound (~0.8GB gather + ~100M f32 atomics per
// iter after dropout), not HBM bound. Total ~2 GFLOP -> WMMA is inapplicable
// (random COO has no dense 16x16xK substructure); we exercise the CDNA5
// async-to-LDS path (global_load_async_to_lds_b128 + s_wait_asynccnt) and
// global_prefetch_b8 instead.

#define PPR_N      100000
#define PPR_E      3200000
#define PPR_C      64
#define PPR_ALPHA  0.15f
#define PPR_KEEP   0.5f
#define PPR_ITERS  5
#define CHUNK      256     // edges per block
#define BLOCK      256     // threads per block (8 wave32)

// ---------------- threefry2x32 (exact JAX semantics) ----------------
__host__ __device__ __forceinline__ unsigned tf_rotl(unsigned x, int r) {
  return (x << r) | (x >> (32 - r));
}

__host__ __device__ __forceinline__ void threefry2x32(unsigned k0, unsigned k1,
                                                      unsigned x0, unsigned x1,
                                                      unsigned& o0, unsigned& o1) {
  const unsigned ks0 = k0, ks1 = k1, ks2 = k0 ^ k1 ^ 0x1BD11BDAu;
  x0 += ks0; x1 += ks1;
#define TF_R(r) { x0 += x1; x1 = tf_rotl(x1, (r)); x1 ^= x0; }
  TF_R(13) TF_R(15) TF_R(26) TF_R(6)   x0 += ks1; x1 += ks2 + 1u;
  TF_R(17) TF_R(29) TF_R(16) TF_R(24)  x0 += ks2; x1 += ks0 + 2u;
  TF_R(13) TF_R(15) TF_R(26) TF_R(6)   x0 += ks0; x1 += ks1 + 3u;
  TF_R(17) TF_R(29) TF_R(16) TF_R(24)  x0 += ks1; x1 += ks2 + 4u;
  TF_R(13) TF_R(15) TF_R(26) TF_R(6)   x0 += ks2; x1 += ks0 + 5u;
#undef TF_R
  o0 = x0; o1 = x1;
}

// random_bits(key, 32, (E,)): counter iota(E) split into halves (x0=[0,E/2), x1=[E/2,E)).
__device__ __forceinline__ unsigned edge_bits(unsigned k0, unsigned k1, unsigned e) {
  const unsigned half = PPR_E / 2u;
  const unsigned j = (e < half) ? e : (e - half);
  unsigned o0, o1;
  threefry2x32(k0, k1, j, j + half, o0, o1);
  return (e < half) ? o0 : o1;
}

// ---------------- CDNA5 async-to-LDS helpers ----------------
// addrspace(3) pointer value == workgroup-relative LDS byte offset, which is
// what global_load_async_to_lds_* expects in its LDS-address VGPR.
__device__ __forceinline__ unsigned lds_off(const void* p) {
  return (unsigned)(unsigned long long)(__attribute__((address_space(3))) const char*)p;
}

__device__ __forceinline__ void async_load_b128(unsigned lds_byte_off, const void* gaddr) {
  // GV mode: per-lane 64-bit global address, per-lane LDS offset, 16B/lane.
  asm volatile("global_load_async_to_lds_b128 %0, %1, off"
               :: "v"(lds_byte_off), "v"(gaddr)
               : "memory");
}

__device__ __forceinline__ void wait_asynccnt0() {
  asm volatile("s_wait_asynccnt 0x0" ::: "memory");
}

// ---------------- kernels (spmm first so the disasm snippet shows it) ----------------
// dst[row] += coef(e) * src[col], coef = keep ? val*(1-a)/keep_prob : 0
__global__ __launch_bounds__(BLOCK) void ppr_spmm(const int* __restrict__ rows,
                                                  const int* __restrict__ cols,
                                                  const float* __restrict__ vals,
                                                  const float* __restrict__ src,
                                                  float* __restrict__ dst,
                                                  unsigned k0, unsigned k1) {
  // One contiguous, 16B-aligned staging buffer: [rows | cols | coef], CHUNK each.
  __shared__ alignas(16) unsigned s_buf[3 * CHUNK];
  int*   s_row  = (int*)s_buf;
  int*   s_col  = (int*)(s_buf + CHUNK);
  float* s_coef = (float*)(s_buf + 2 * CHUNK);

  const int t    = threadIdx.x;
  const int base = blockIdx.x * CHUNK;

  // Stage this block's edge metadata via the CDNA5 async-to-LDS path.
  // 3 arrays x 1KB; threads 0..191 each move 16B (4 dwords) asynchronously.
  {
    const int g = t >> 6;          // 64-thread (2-wave) groups, uniform per wave
    const int u = (t & 63) * 4;    // dword index within array
    if (g == 0) {
      async_load_b128(lds_off(&s_row[u]),  (const void*)(rows + base + u));
    } else if (g == 1) {
      async_load_b128(lds_off(&s_col[u]),  (const void*)(cols + base + u));
    } else if (g == 2) {
      async_load_b128(lds_off(&s_coef[u]), (const void*)(vals + base + u));
    }
  }
  wait_asynccnt0();   // each wave drains its own ASYNCcnt ...
  __syncthreads();    // ... then the barrier makes all staged data visible

  // Phase 1: exact-JAX dropout -> per-edge coefficient (thread t owns slot t).
  {
    const unsigned bits = edge_bits(k0, k1, (unsigned)(base + t));
    const float u = __uint_as_float((bits >> 9) | 0x3f800000u) - 1.0f; // [0,1)
    const float scale = (1.0f - PPR_ALPHA) / PPR_KEEP;
    const float coef = (u < PPR_KEEP) ? s_coef[t] * scale : 0.0f;
    s_coef[t] = coef;
    if (coef != 0.0f) {
      // Warm caches for the row this block will gather (global_prefetch_b8).
      __builtin_prefetch(&src[(size_t)s_col[t] * PPR_C], 0, 3);
    }
  }
  __syncthreads();

  // Phase 2: 16 lanes x float4 cover one 64-wide row; 16 edges in flight/pass.
  const int lane = t & 15;   // channel group: columns [lane*4, lane*4+4)
  const int eg   = t >> 4;   // edge sub-slot 0..15
  #pragma unroll
  for (int k = 0; k < CHUNK / 16; ++k) {
    const int   el = k * 16 + eg;
    const float cf = s_coef[el];
    if (cf != 0.0f) {                 // dropped edges: no gather, no atomics
      const int r = s_row[el];
      const int c = s_col[el];
      const float4 h = ((const float4*)(src + (size_t)c * PPR_C))[lane]; // global_load_b128
      float* dp = dst + (size_t)r * PPR_C + lane * 4;
      atomicAdd(dp + 0, cf * h.x);    // global_atomic_add_f32 (L2-resident dst)
      atomicAdd(dp + 1, cf * h.y);
      atomicAdd(dp + 2, cf * h.z);
      atomicAdd(dp + 3, cf * h.w);
    }
  }
}

// dst = alpha * H0  (folds the affine epilogue into the accumulator init)
__global__ __launch_bounds__(BLOCK) void ppr_init(const float* __restrict__ h0,
                                                  float* __restrict__ dst) {
  const int i = blockIdx.x * BLOCK + threadIdx.x;      // float4 index
  float4 v = ((const float4*)h0)[i];
  v.x *= PPR_ALPHA; v.y *= PPR_ALPHA; v.z *= PPR_ALPHA; v.w *= PPR_ALPHA;
  ((float4*)dst)[i] = v;
}

// ---------------- host orchestration ----------------
extern "C" void kernel_launch(void* const* d_in, const int* in_sizes, int n_in,
                              void* d_out, int out_size, void* d_ws, size_t ws_size,
                              hipStream_t stream) {
  (void)in_sizes; (void)n_in; (void)out_size; (void)ws_size;

  const float* logits = (const float*)d_in[0];
  const int*   eidx   = (const int*)d_in[1];     // [2, E] row-major
  const float* evals  = (const float*)d_in[2];
  const int*   rows   = eidx;
  const int*   cols   = eidx + PPR_E;

  float* out = (float*)d_out;
  float* ws  = (float*)d_ws;                     // needs N*C*4 = 25.6 MB

  // Host-side: keys = jax.random.split(jax.random.key(42), 5)
  //   split = random_bits(key,32,(10,)) with counter iota(10) halved.
  unsigned b[10];
  for (int j = 0; j < 5; ++j) {
    unsigned o0, o1;
    threefry2x32(0u, 42u, (unsigned)j, (unsigned)(j + 5), o0, o1);
    b[j] = o0; b[j + 5] = o1;
  }

  const int initGrid = (PPR_N * PPR_C / 4) / BLOCK;  // 6250 exact
  const int spmmGrid = PPR_E / CHUNK;                // 12500 exact

  const float* src = logits;
  float* bufs[2] = { out, ws };                  // dsts: out, ws, out, ws, out
  for (int it = 0; it < PPR_ITERS; ++it) {
    float* dstb = bufs[it & 1];
    ppr_init<<<initGrid, BLOCK, 0, stream>>>(logits, dstb);
    ppr_spmm<<<spmmGrid, BLOCK, 0, stream>>>(rows, cols, evals, src, dstb,
                                             b[2 * it], b[2 * it + 1]);
    src = dstb;
  }
}